// PSROIPool_11982958756198
// MI455X (gfx1250) — compile-verified
//
#include <hip/hip_runtime.h>

// PS-ROI average pooling for MI455X (gfx1250), direct-reduction formulation.
// One block per ROI; 8 wave32 waves; 49 groups x 16 bins per ROI.
// Bin sums computed with V_WMMA_F32_16X16X4_F32 as a wave-wide 64-MAC
// accumulator (B = ones). Two interleaved accumulators break the C->D chain;
// loads are branchless (clamped index + cndmask); row/col indexing uses a
// pow2-padded bin width (shift/and); the chunk loop trip count is forced
// into an SGPR via readfirstlane so EXEC provably stays all-1s for WMMA.

typedef __attribute__((ext_vector_type(2))) float v2f;
typedef __attribute__((ext_vector_type(8))) float v8f;

#define GS_   7
#define D_    16
#define PH_   7
#define PW_   7
#define Hf    96
#define Wf    96
#define Cf    (D_ * GS_ * GS_)   /* 784 */
#define SCALE_ 0.0625f

__global__ __launch_bounds__(256) void psroi_wmma_kernel(
    const float* __restrict__ feat,   // [N, 784, 96, 96]
    const float* __restrict__ rois,   // [R, 5]
    float* __restrict__ out,          // [R, 784]
    int totalm1)                      // last valid flat index into feat
{
    __shared__ int s_hs[PH_], s_he[PH_], s_ws[PW_], s_we[PW_];
    __shared__ int s_cnt[PH_ * PW_];
    __shared__ int s_b;

    const int roi = blockIdx.x;
    const int tid = threadIdx.x;

    // --- Per-ROI geometry, computed once into LDS (threads 0..48).
    // Each thread recomputes from the ROI row directly: no staging barrier.
    if (tid < PH_ * PW_) {
        const float r0 = rois[roi * 5 + 0];
        const float x1 = rintf(rois[roi * 5 + 1]) * SCALE_;
        const float y1 = rintf(rois[roi * 5 + 2]) * SCALE_;
        const float x2 = rintf(rois[roi * 5 + 3] + 1.0f) * SCALE_;
        const float y2 = rintf(rois[roi * 5 + 4] + 1.0f) * SCALE_;
        const float roi_w = fmaxf(x2 - x1, 0.1f);
        const float roi_h = fmaxf(y2 - y1, 0.1f);
        const float bh = roi_h / (float)PH_;
        const float bw = roi_w / (float)PW_;
        const int   ph = tid / PW_;
        const int   pw = tid - ph * PW_;
        const float fh = (float)ph;
        const float fw = (float)pw;
        const int hs = (int)fminf(fmaxf(floorf(fh * bh + y1),         0.0f), (float)Hf);
        const int he = (int)fminf(fmaxf(ceilf((fh + 1.0f) * bh + y1), 0.0f), (float)Hf);
        const int ws = (int)fminf(fmaxf(floorf(fw * bw + x1),         0.0f), (float)Wf);
        const int we = (int)fminf(fmaxf(ceilf((fw + 1.0f) * bw + x1), 0.0f), (float)Wf);
        s_cnt[tid] = (he - hs) * (we - ws);
        if (pw == 0) { s_hs[ph] = hs; s_he[ph] = he; }   // row geometry (pw-independent)
        if (ph == 0) { s_ws[pw] = ws; s_we[pw] = we; }   // col geometry (ph-independent)
        if (tid == 0) s_b = (int)r0;
    }
    __syncthreads();

    const int lane = tid & 31;
    const int wave = tid >> 5;
    const int m    = lane & 15;   // A-matrix row (bin within group)
    const int half = lane >> 4;   // 0: supplies K=4t+{0,1}; 1: K=4t+{2,3}
    const int b    = s_b;

    const v2f Bones = {1.0f, 1.0f};   // B = ones(4x16): D[m,n] += sum_k A[m,k]

    for (int g = wave; g < 49; g += 8) {          // 49*16 == 784 bins, exact
        const int c   = g * 16 + m;               // bin index == channel index
        const int dd  = c / 49;
        const int rem = c - dd * 49;
        const int ph  = rem / 7;
        const int pw  = rem - ph * 7;

        const int hs = s_hs[ph], he = s_he[ph];
        const int ws = s_ws[pw], we = s_we[pw];
        const int rh = he - hs, rw = we - ws;

        // pow2-padded width: row = e >> shift, col = e & (rwP-1)
        const int shift = (rw <= 1) ? 0 : (32 - __clz(rw - 1));
        const int colMask = (1 << shift) - 1;
        const int areaP = rh << shift;            // padded stream length

        const int baseIdx = ((b * Cf + c) * Hf + hs) * Wf + ws;
        __builtin_prefetch(feat + min(baseIdx, totalm1), 0, 1); // global_prefetch_b8

        // Uniform trip count: wave-max of per-bin chunk counts, rounded to 2,
        // then pinned to an SGPR so the loop is scalar and EXEC stays full.
        int steps = (areaP + 3) >> 2;
        #pragma unroll
        for (int off = 16; off; off >>= 1)
            steps = max(steps, __shfl_xor(steps, off, 32));
        const int steps2 = __builtin_amdgcn_readfirstlane((steps + 1) & ~1);

        v8f acc0 = {0.f,0.f,0.f,0.f,0.f,0.f,0.f,0.f};
        v8f acc1 = {0.f,0.f,0.f,0.f,0.f,0.f,0.f,0.f};

        for (int t = 0; t < steps2; t += 2) {
            // ---- chunk t -> acc0 (branchless masked loads) ----
            const int eA0 = 4 * t + 2 * half;
            const int rA0 = eA0 >> shift,  cA0 = eA0 & colMask;
            const int rA1 = (eA0+1) >> shift, cA1 = (eA0+1) & colMask;
            const bool vA0 = (rA0 < rh) & (cA0 < rw);
            const bool vA1 = (rA1 < rh) & (cA1 < rw);
            float a0 = feat[min(baseIdx + rA0 * Wf + cA0, totalm1)];
            float a1 = feat[min(baseIdx + rA1 * Wf + cA1, totalm1)];
            a0 = vA0 ? a0 : 0.0f;
            a1 = vA1 ? a1 : 0.0f;

            // ---- chunk t+1 -> acc1 ----
            const int eB0 = 4 * (t + 1) + 2 * half;
            const int rB0 = eB0 >> shift,  cB0 = eB0 & colMask;
            const int rB1 = (eB0+1) >> shift, cB1 = (eB0+1) & colMask;
            const bool vB0 = (rB0 < rh) & (cB0 < rw);
            const bool vB1 = (rB1 < rh) & (cB1 < rw);
            float b0 = feat[min(baseIdx + rB0 * Wf + cB0, totalm1)];
            float b1 = feat[min(baseIdx + rB1 * Wf + cB1, totalm1)];
            b0 = vB0 ? b0 : 0.0f;
            b1 = vB1 ? b1 : 0.0f;

            const v2f A = {a0, a1};
            const v2f B = {b0, b1};
            acc0 = __builtin_amdgcn_wmma_f32_16x16x4_f32(
                false, A, false, Bones, (short)0, acc0, false, false);
            acc1 = __builtin_amdgcn_wmma_f32_16x16x4_f32(
                false, B, false, Bones, (short)0, acc1, false, false);
        }

        // D layout: VGPR r, lanes 0-15 -> M=r; lanes 16-31 -> M=r+8.
        // Lane 0 writes bins 0..7 of the group, lane 16 writes bins 8..15.
        if (m == 0) {
#pragma unroll
            for (int r = 0; r < 8; ++r) {
                const int cc   = g * 16 + half * 8 + r;
                const int d2   = cc / 49;
                const int rem2 = cc - d2 * 49;
                const int cnt  = s_cnt[rem2];
                const float s  = acc0[r] + acc1[r];
                out[roi * Cf + cc] = (cnt > 0) ? (s / (float)cnt) : 0.0f;
            }
        }
    }
}

extern "C" void kernel_launch(void* const* d_in, const int* in_sizes, int n_in,
                              void* d_out, int out_size, void* d_ws, size_t ws_size,
                              hipStream_t stream) {
    const float* feat = (const float*)d_in[0];
    const float* rois = (const float*)d_in[1];
    float* out = (float*)d_out;
    const int R = in_sizes[1] / 5;        // rois rows
    const int totalm1 = in_sizes[0] - 1;  // last valid flat feature index
    psroi_wmma_kernel<<<R, 256, 0, stream>>>(feat, rois, out, totalm1);
}